// LaplaceMeshLoss_17154099381029
// MI455X (gfx1250) — compile-verified
//
#include <hip/hip_runtime.h>

// -------- CDNA5 vector types --------
typedef __attribute__((ext_vector_type(2))) float v2f;
typedef __attribute__((ext_vector_type(8))) float v8f;

#define N_MESHES 4

__device__ __forceinline__ void fatom_add(float* p, float v) {
    // Expect global_atomic_add_f32 (relaxed, device scope, accumulators are L2-resident)
    __hip_atomic_fetch_add(p, v, __ATOMIC_RELAXED, __HIP_MEMORY_SCOPE_AGENT);
}

// ---------------------------------------------------------------------------
// Pass 1: COO scatter.  Lx/Ly/Lz/rsum are SoA accumulators in d_ws (zeroed).
// Each edge: 12B coalesced stream + 12B L2 gather + 4 f32 atomics (L2 RMW).
// ---------------------------------------------------------------------------
__global__ void lml_scatter(const float* __restrict__ verts,
                            const float* __restrict__ vals,
                            const int*   __restrict__ rows,
                            const int*   __restrict__ cols,
                            float* __restrict__ Lx, float* __restrict__ Ly,
                            float* __restrict__ Lz, float* __restrict__ rsum,
                            int nnz) {
    const int stride = gridDim.x * blockDim.x;
    for (int e = blockIdx.x * blockDim.x + threadIdx.x; e < nnz; e += stride) {
        if (e + stride < nnz) {
            // global_prefetch_b8: pull next iteration's stream into cache while
            // this iteration's gather + atomics are in flight.
            __builtin_prefetch(&vals[e + stride], 0, 0);
            __builtin_prefetch(&cols[e + stride], 0, 0);
            __builtin_prefetch(&rows[e + stride], 0, 0);
        }
        const float v = vals[e];
        const int   c = cols[e];
        const int   r = rows[e];
        const float* vp = verts + 3ull * (unsigned)c;
        const float x = vp[0], y = vp[1], z = vp[2];
        fatom_add(&Lx[r],   v * x);
        fatom_add(&Ly[r],   v * y);
        fatom_add(&Lz[r],   v * z);
        fatom_add(&rsum[r], v);
    }
}

// ---------------------------------------------------------------------------
// Pass 2: vertices-per-mesh histogram (LDS counters, 4 global atomics/block).
// ---------------------------------------------------------------------------
__global__ void lml_mesh_hist(const int* __restrict__ midx, int* __restrict__ nvpm,
                              int V) {
    __shared__ int cnt[N_MESHES];
    if (threadIdx.x < N_MESHES) cnt[threadIdx.x] = 0;
    __syncthreads();
    const int stride = gridDim.x * blockDim.x;
    for (int i = blockIdx.x * blockDim.x + threadIdx.x; i < V; i += stride) {
        atomicAdd(&cnt[midx[i]], 1);
    }
    __syncthreads();
    if (threadIdx.x < N_MESHES) atomicAdd(&nvpm[threadIdx.x], cnt[threadIdx.x]);
}

// ---------------------------------------------------------------------------
// Pass 3: per-vertex loss + reduction.
// Wave reduction via V_WMMA_F32_16X16X4_F32:
//   A (16x4 f32) = lane partials  (lane L: A[L%16][2*(L/16)] = acc)
//   B (4x16 f32) = all ones  ->  D[i][j] = rowsum_i  (= acc_i + acc_{i+16})
//   lanes 0-15 hold rows 0-7 in c[0..7]; lanes 16-31 hold rows 8-15.
// One v_wmma + 7 adds + 1 shfl collapses 32 lanes in full f32 precision.
// Trip count of the grid-stride loop is uniform, so EXEC is all-1s at the WMMA.
// ---------------------------------------------------------------------------
__global__ void lml_finalize(const float* __restrict__ verts,
                             const float* __restrict__ Lx,
                             const float* __restrict__ Ly,
                             const float* __restrict__ Lz,
                             const float* __restrict__ rsum,
                             const int*   __restrict__ midx,
                             const float* __restrict__ coefs,
                             const int*   __restrict__ nvpm,
                             float* __restrict__ out,
                             int V) {
    const int tid    = blockIdx.x * blockDim.x + threadIdx.x;
    const int stride = gridDim.x * blockDim.x;

    float acc = 0.0f;
    // Uniform trip count: every thread executes the same number of iterations.
    for (int base = 0; base < V; base += stride) {
        const int i = base + tid;
        if (i < V) {
            const float rs = rsum[i];
            const float nw = (rs > 0.0f) ? (1.0f / rs) : rs;
            const float* vp = verts + 3ull * (unsigned)i;
            const float rx = Lx[i] * nw - vp[0];
            const float ry = Ly[i] * nw - vp[1];
            const float rz = Lz[i] * nw - vp[2];
            const float l  = sqrtf(rx * rx + ry * ry + rz * rz);
            const float w  = 1.0f / (float)nvpm[midx[i]];
            acc += l * w * coefs[i];
        }
    }

    // ---- WMMA wave32 reduction (f32, 16x16x4) ----
    v2f a; a.x = acc;  a.y = 0.0f;
    v2f b; b.x = 1.0f; b.y = 1.0f;
    v8f c = {};
    c = __builtin_amdgcn_wmma_f32_16x16x4_f32(
            /*neg_a=*/false, a, /*neg_b=*/false, b,
            /*c_mod=*/(short)0, c, /*reuse_a=*/false, /*reuse_b=*/false);
    float s = c[0] + c[1] + c[2] + c[3] + c[4] + c[5] + c[6] + c[7];
    // lanes 0-15: rows 0-7 total; lanes 16-31: rows 8-15 total. Combine halves.
    s += __shfl_xor(s, 16, 32);
    if ((threadIdx.x & 31) == 0) {
        fatom_add(out, s * (1.0f / (float)N_MESHES));
    }
}

// ---------------------------------------------------------------------------
// Launcher
// ---------------------------------------------------------------------------
extern "C" void kernel_launch(void* const* d_in, const int* in_sizes, int n_in,
                              void* d_out, int out_size, void* d_ws, size_t ws_size,
                              hipStream_t stream) {
    const float* verts = (const float*)d_in[0];  // [V,3]
    const float* vals  = (const float*)d_in[1];  // [NNZ]
    const int*   rows  = (const int*)  d_in[2];  // [NNZ]
    const int*   cols  = (const int*)  d_in[3];  // [NNZ]
    const int*   midx  = (const int*)  d_in[4];  // [V]
    const float* coefs = (const float*)d_in[5];  // [V]

    const int V   = in_sizes[0] / 3;
    const int NNZ = in_sizes[1];

    // Workspace layout (SoA accumulators, then nvpm)
    float* w   = (float*)d_ws;
    float* Lx  = w;
    float* Ly  = w + (size_t)V;
    float* Lz  = w + 2ull * V;
    float* rs  = w + 3ull * V;
    int*  nvpm = (int*)(w + 4ull * V);

    const size_t ws_used = (size_t)4 * V * sizeof(float) + N_MESHES * sizeof(int);
    hipMemsetAsync(d_ws, 0, ws_used, stream);
    hipMemsetAsync(d_out, 0, (size_t)out_size * sizeof(float), stream);

    const int TPB = 256;
    int blocks_sc = (NNZ + TPB - 1) / TPB;
    if (blocks_sc > 16384) blocks_sc = 16384;
    lml_scatter<<<blocks_sc, TPB, 0, stream>>>(verts, vals, rows, cols,
                                               Lx, Ly, Lz, rs, NNZ);

    int blocks_h = (V + TPB - 1) / TPB;
    if (blocks_h > 2048) blocks_h = 2048;
    lml_mesh_hist<<<blocks_h, TPB, 0, stream>>>(midx, nvpm, V);

    int blocks_f = (V + TPB - 1) / TPB;
    if (blocks_f > 4096) blocks_f = 4096;
    lml_finalize<<<blocks_f, TPB, 0, stream>>>(verts, Lx, Ly, Lz, rs,
                                               midx, coefs, nvpm,
                                               (float*)d_out, V);
}